// CGElementProduct_16243566313948
// MI455X (gfx1250) — compile-verified
//
#include <hip/hip_runtime.h>
#include <cstddef>

typedef float v2f __attribute__((ext_vector_type(2)));
typedef float v8f __attribute__((ext_vector_type(8)));

// ---------------- compile-time Clebsch-Gordan tables (mirror reference _cg) ----------------
constexpr double factd(int n){ double r=1.0; for(int i=2;i<=n;++i) r*=(double)i; return r; }
constexpr double csqrt_(double x){ if(x<=0.0) return 0.0; double r=(x>1.0)?x:1.0;
  for(int i=0;i<64;++i) r=0.5*(r+x/r); return r; }
constexpr double cgc(int j1,int m1,int j2,int m2,int J,int M){
  if(m1+m2!=M) return 0.0;
  int adj=j1-j2; if(adj<0) adj=-adj;
  if(J<adj || J>j1+j2) return 0.0;
  if(m1<-j1||m1>j1||m2<-j2||m2>j2||M<-J||M>J) return 0.0;
  double pref = csqrt_((2.0*J+1.0)*factd(J+j1-j2)*factd(J-j1+j2)*factd(j1+j2-J)/factd(j1+j2+J+1));
  pref *= csqrt_(factd(J+M)*factd(J-M)*factd(j1-m1)*factd(j1+m1)*factd(j2-m2)*factd(j2+m2));
  double s=0.0;
  int kmin=0; if(j2-J-m1>kmin)kmin=j2-J-m1; if(j1+m2-J>kmin)kmin=j1+m2-J;
  int kmax=j1+j2-J; if(j1-m1<kmax)kmax=j1-m1; if(j2+m2<kmax)kmax=j2+m2;
  for(int k=kmin;k<=kmax;++k){
    double t=1.0/(factd(k)*factd(j1+j2-J-k)*factd(j1-m1-k)*factd(j2+m2-k)
                  *factd(J-j1-m2+k)*factd(J-j2+m1+k));
    s += (k&1)? -t : t;
  }
  return pref*s;
}

struct Tab1 { float v[3][5][3]; };            // [ll][M+ll][m+1]
constexpr Tab1 mk1(){
  Tab1 t{};
  for(int ll=0;ll<3;++ll) for(int Mi=0;Mi<2*ll+1;++Mi) for(int mi=0;mi<3;++mi){
    int m=mi-1, M=Mi-ll, u=M-m;
    t.v[ll][Mi][mi] = (u<-1||u>1)?0.0f:(float)cgc(1,m,1,u,ll,M);
  }
  return t;
}
constexpr Tab1 T1 = mk1();

struct Tab2 { float v[5][9][5]; };            // [ll][M+ll][m+2]
constexpr Tab2 mk2(){
  Tab2 t{};
  for(int ll=0;ll<5;++ll) for(int Mi=0;Mi<2*ll+1;++Mi) for(int mi=0;mi<5;++mi){
    int m=mi-2, M=Mi-ll, u=M-m;
    t.v[ll][Mi][mi] = (u<-2||u>2)?0.0f:(float)cgc(2,m,2,u,ll,M);
  }
  return t;
}
constexpr Tab2 T2 = mk2();

// ---------------- WMMA f32 16x16x4 ----------------
__device__ __forceinline__ v8f wmma4(v2f a, v2f b, v8f c){
  return __builtin_amdgcn_wmma_f32_16x16x4_f32(false, a, false, b, (short)0, c, false, false);
}

// W[b][k][Mi] = c(L,LL,k,Mi) * x[b][Mi-k+2L-LL]; all indices compile-time so the
// CG constant folds to an immediate and x[] stays in registers.
template<int L, int LL, int Mi, int K>
__device__ __forceinline__ float wv(const float* x){
  constexpr int ui = Mi - K + 2*L - LL;
  constexpr bool ok = (K <= 2*L) && (ui >= 0) && (ui <= 2*L);
  constexpr int LL1 = (LL < 3) ? LL : 0;
  constexpr int Mi1 = (Mi < 5) ? Mi : 0;
  constexpr int K1  = (K  < 3) ? K  : 0;
  constexpr int K2  = (K  < 5) ? K  : 0;
  constexpr float c = !ok ? 0.0f : (L==1 ? T1.v[LL1][Mi1][K1] : T2.v[LL][Mi][K2]);
  constexpr int uic = ok ? ui : 0;
  if(!ok) return 0.0f;
  return c * x[uic];
}

template<int L, int LL, int Mi>
__device__ __forceinline__ void compute_one_m(
    const float* xr, const float* xi, bool hi,
    v2f Ar0, v2f Ai0, v2f An0, v2f Ar1, v2f Ai1, v2f An1,
    v8f& cR, v8f& cI)
{
  // B fragment chunk0: lanes 0-15 hold k=0,1; lanes 16-31 hold k=2,3
  v2f Br0, Bi0;
  Br0.x = hi ? wv<L,LL,Mi,2>(xr) : wv<L,LL,Mi,0>(xr);
  Br0.y = hi ? wv<L,LL,Mi,3>(xr) : wv<L,LL,Mi,1>(xr);
  Bi0.x = hi ? wv<L,LL,Mi,2>(xi) : wv<L,LL,Mi,0>(xi);
  Bi0.y = hi ? wv<L,LL,Mi,3>(xi) : wv<L,LL,Mi,1>(xi);
  v8f z = {0.f,0.f,0.f,0.f,0.f,0.f,0.f,0.f};
  cR = z; cI = z;
  if (L == 2){
    // chunk1: k=4..7 (only k=4 is live)
    v2f Br1, Bi1;
    Br1.x = hi ? 0.0f : wv<L,LL,Mi,4>(xr);  Br1.y = 0.0f;
    Bi1.x = hi ? 0.0f : wv<L,LL,Mi,4>(xi);  Bi1.y = 0.0f;
    cR = wmma4(An1, Bi1, cR);   // -Xi @ Wi
    cR = wmma4(Ar1, Br1, cR);   //  Xr @ Wr
    cI = wmma4(Ai1, Br1, cI);   //  Xi @ Wr
    cI = wmma4(Ar1, Bi1, cI);   //  Xr @ Wi
  }
  cR = wmma4(An0, Bi0, cR);
  cR = wmma4(Ar0, Br0, cR);
  cI = wmma4(Ai0, Br0, cI);
  cI = wmma4(Ar0, Bi0, cI);
}

template<int L, int LL, int Mi, int NM>
struct MRec {
  static __device__ __forceinline__ void run(
      const float* xr, const float* xi, bool hi,
      v2f Ar0, v2f Ai0, v2f An0, v2f Ar1, v2f Ai1, v2f An1,
      v8f* accR, v8f* accI){
    compute_one_m<L,LL,Mi>(xr,xi,hi,Ar0,Ai0,An0,Ar1,Ai1,An1,accR[Mi],accI[Mi]);
    MRec<L,LL,Mi+1,NM>::run(xr,xi,hi,Ar0,Ai0,An0,Ar1,Ai1,An1,accR,accI);
  }
};
template<int L, int LL, int NM>
struct MRec<L,LL,NM,NM> {
  static __device__ __forceinline__ void run(
      const float*, const float*, bool, v2f,v2f,v2f,v2f,v2f,v2f, v8f*, v8f*){}
};

// One (l,ll) block: all 2*LL+1 WMMA accumulator pairs kept live for the norm.
template<int L, int LL>
__device__ __forceinline__ void pair_block(
    const float* xr, const float* xi, bool hi, int r,
    v2f Ar0, v2f Ai0, v2f An0, v2f Ar1, v2f Ai1, v2f An1, float* yout)
{
  constexpr int NM = 2*LL+1;
  v8f accR[NM], accI[NM];
  MRec<L,LL,0,NM>::run(xr,xi,hi,Ar0,Ai0,An0,Ar1,Ai1,An1,accR,accI);

  constexpr int base = (LL==0)?0:(LL==1)?288:(LL==2)?1056:(LL==3)?2496:3336;
  constexpr int t0 = (L==1) ? ((LL==2)?0:16)
                            : ((LL==0)?152:(LL==1)?136:(LL==2)?152:0);
  const int abase = hi ? 8 : 0;
  #pragma unroll
  for(int ri=0; ri<8; ++ri){
    const int a = abase + ri;          // matrix row = first channel index
    const int b = r;                   // matrix col = second channel index
    const bool keep = (a<b) || ((a==b) && ((LL&1)==0));
    if(keep){
      float nrm = 0.0f;
      #pragma unroll
      for(int Mi=0;Mi<NM;++Mi){
        float yr = accR[Mi][ri], yi2 = accI[Mi][ri];
        nrm += yr*yr + yi2*yi2;
      }
      // scale = 1/(eps + nrm^(1/4)) with nrm >= 0: use native v_sqrt_f32 + v_rcp_f32
      // (3 VALU ops instead of the ~25-op IEEE sqrt/div expansions)
      const float q  = __builtin_amdgcn_sqrtf(__builtin_amdgcn_sqrtf(nrm));
      const float sc = __builtin_amdgcn_rcpf(1e-5f + q);
      const int tseg = ((LL&1)==0) ? ((a*(31-a))>>1) + b
                                   : ((a*(29-a))>>1) + b - 1;
      const int ch0 = base + (t0 + tseg)*NM;
      #pragma unroll
      for(int Mi=0;Mi<NM;++Mi){
        float2 v; v.x = accR[Mi][ri]*sc; v.y = accI[Mi][ri]*sc;
        *reinterpret_cast<float2*>(yout + (size_t)(ch0+Mi)*6400) = v;
      }
    }
  }
}

// One wave per site (b,t,f). 12800 sites = 1600 blocks x 8 waves, all waves full.
__global__ __launch_bounds__(256)
void cg_tensor_product_wmma(const float* __restrict__ x, float* __restrict__ y)
{
  const int lane = (int)(threadIdx.x & 31u);
  const int wid  = (int)blockIdx.x * 8 + (int)(threadIdx.x >> 5);
  const int ff = wid & 63;
  const int bt = wid >> 6;
  const int tt = bt % 50;
  const int bi = bt / 50;
  const size_t px = (size_t)(tt*64+ff)*2;
  const float* xin  = x + (size_t)bi*144*6400  + px;   // channel stride = T*F*2 = 6400
  float*       yout = y + (size_t)bi*4560*6400 + px;
  const int  r  = lane & 15;
  const bool hi = lane >= 16;

  // l=0 pass-through: out ch 0..15 <- in ch 0..15
  if(lane < 16){
    float2 v = *reinterpret_cast<const float2*>(xin + (size_t)lane*6400);
    *reinterpret_cast<float2*>(yout + (size_t)lane*6400) = v;
  }

  v2f Z; Z.x = 0.0f; Z.y = 0.0f;

  // ---------------- L = 1 (K=3, one WMMA k-chunk) ----------------
  {
    float xr[3], xi_[3];
    #pragma unroll
    for(int u=0;u<3;++u){
      float2 v = *reinterpret_cast<const float2*>(xin + (size_t)(16 + r*3 + u)*6400);
      xr[u]=v.x; xi_[u]=v.y;
    }
    if(!hi){ // pass-through l=1: out ch 288 + r*3 + u
      #pragma unroll
      for(int u=0;u<3;++u){
        float2 v; v.x=xr[u]; v.y=xi_[u];
        *reinterpret_cast<float2*>(yout + (size_t)(288 + r*3 + u)*6400) = v;
      }
    }
    v2f Ar0, Ai0, An0;
    Ar0.x = hi ? xr[2]  : xr[0];   Ar0.y = hi ? 0.0f : xr[1];
    Ai0.x = hi ? xi_[2] : xi_[0];  Ai0.y = hi ? 0.0f : xi_[1];
    An0 = -Ai0;
    pair_block<1,0>(xr, xi_, hi, r, Ar0, Ai0, An0, Z, Z, Z, yout);
    pair_block<1,1>(xr, xi_, hi, r, Ar0, Ai0, An0, Z, Z, Z, yout);
    pair_block<1,2>(xr, xi_, hi, r, Ar0, Ai0, An0, Z, Z, Z, yout);
  }

  // ---------------- L = 2 (K=5, two WMMA k-chunks) ----------------
  {
    float xr[5], xi_[5];
    #pragma unroll
    for(int u=0;u<5;++u){
      float2 v = *reinterpret_cast<const float2*>(xin + (size_t)(64 + r*5 + u)*6400);
      xr[u]=v.x; xi_[u]=v.y;
    }
    if(!hi){ // pass-through l=2: out ch 1056 + (136+r)*5 + u
      #pragma unroll
      for(int u=0;u<5;++u){
        float2 v; v.x=xr[u]; v.y=xi_[u];
        *reinterpret_cast<float2*>(yout + (size_t)(1056 + (136 + r)*5 + u)*6400) = v;
      }
    }
    v2f Ar0, Ai0, An0, Ar1, Ai1, An1;
    Ar0.x = hi ? xr[2]  : xr[0];   Ar0.y = hi ? xr[3]  : xr[1];
    Ai0.x = hi ? xi_[2] : xi_[0];  Ai0.y = hi ? xi_[3] : xi_[1];
    Ar1.x = hi ? 0.0f   : xr[4];   Ar1.y = 0.0f;
    Ai1.x = hi ? 0.0f   : xi_[4];  Ai1.y = 0.0f;
    An0 = -Ai0; An1 = -Ai1;
    pair_block<2,0>(xr, xi_, hi, r, Ar0, Ai0, An0, Ar1, Ai1, An1, yout);
    pair_block<2,1>(xr, xi_, hi, r, Ar0, Ai0, An0, Ar1, Ai1, An1, yout);
    pair_block<2,2>(xr, xi_, hi, r, Ar0, Ai0, An0, Ar1, Ai1, An1, yout);
    pair_block<2,3>(xr, xi_, hi, r, Ar0, Ai0, An0, Ar1, Ai1, An1, yout);
    pair_block<2,4>(xr, xi_, hi, r, Ar0, Ai0, An0, Ar1, Ai1, An1, yout);
  }
}

extern "C" void kernel_launch(void* const* d_in, const int* in_sizes, int n_in,
                              void* d_out, int out_size, void* d_ws, size_t ws_size,
                              hipStream_t stream)
{
  (void)in_sizes; (void)n_in; (void)out_size; (void)d_ws; (void)ws_size;
  const float* x = (const float*)d_in[0];
  float* y = (float*)d_out;
  // 12800 sites, 1 wave/site, 8 waves/block -> 1600 blocks of 256 threads
  cg_tensor_product_wmma<<<dim3(1600), dim3(256), 0, stream>>>(x, y);
}